// GeneralPulling_72808285602046
// MI455X (gfx1250) — compile-verified
//
#include <hip/hip_runtime.h>
#include <hip/hip_bf16.h>

#define ALPHA_F 1.0f
#define NB      1024     // B
#define NBITS   128
#define KSTEPS  (NBITS / 4)

typedef float v2f __attribute__((ext_vector_type(2)));
typedef float v4f __attribute__((ext_vector_type(4)));
typedef float v8f __attribute__((ext_vector_type(8)));

// ---------------------------------------------------------------------------
// Kernel 0: row squared-norms of embeddings (sqA) and ref_embeddings (sqB).
// 2048 threads total, one row per thread, float4 reads.
// ---------------------------------------------------------------------------
__global__ void gp_norms_kernel(const float* __restrict__ emb,
                                const float* __restrict__ ref,
                                float* __restrict__ sqA,
                                float* __restrict__ sqB) {
    int i = blockIdx.x * blockDim.x + threadIdx.x;
    if (i >= 2 * NB) return;
    const float* row = (i < NB) ? (emb + (size_t)i * NBITS)
                                : (ref + (size_t)(i - NB) * NBITS);
    float s = 0.f;
    #pragma unroll 4
    for (int c = 0; c < NBITS; c += 4) {
        v4f v = *(const v4f*)(row + c);
        s += v.x * v.x + v.y * v.y + v.z * v.z + v.w * v.w;
    }
    if (i < NB) sqA[i] = s; else sqB[i - NB] = s;
}

// ---------------------------------------------------------------------------
// Kernel 1: dist[m][n] = max(sqrt(max(sqA[m]+sqB[n]-2*dot, 0)), 1e-6)
// One 16x16 output tile per wave; K-loop of 32 v_wmma_f32_16x16x4_f32.
//
// fp32 WMMA per-lane layout (ISA 7.12.2):
//   A (16x4):  lane l in [0,16), half h: VGPR0=K(2h), VGPR1=K(2h+1), row M=l
//   B (4x16):  same striping with column N=l
//   C/D (16x16 f32, 8 VGPRs): element [M = v + 8*h][N = l] in VGPR v, lane 16h+l
// ---------------------------------------------------------------------------
__global__ void gp_dist_wmma_kernel(const float* __restrict__ emb,
                                    const float* __restrict__ ref,
                                    const float* __restrict__ sqA,
                                    const float* __restrict__ sqB,
                                    float* __restrict__ dist) {
    const int wave   = threadIdx.x >> 5;            // 8 waves / block
    const int lane   = threadIdx.x & 31;
    const int half   = lane >> 4;                   // 0 or 1
    const int l      = lane & 15;
    const int tile   = blockIdx.x * 8 + wave;       // 4096 tiles total
    const int m0     = (tile >> 6) << 4;            // tile row * 16
    const int n0     = (tile & 63) << 4;            // tile col * 16

    const float* arow = emb + (size_t)(m0 + l) * NBITS + 2 * half;
    const float* brow = ref + (size_t)(n0 + l) * NBITS + 2 * half;

    v8f acc = {};
    #pragma unroll 8
    for (int k = 0; k < KSTEPS; ++k) {
        v2f a = *(const v2f*)(arow + 4 * k);
        v2f b = *(const v2f*)(brow + 4 * k);
        acc = __builtin_amdgcn_wmma_f32_16x16x4_f32(
            /*neg_a=*/false, a, /*neg_b=*/false, b,
            /*c_mod=*/(short)0, acc, /*reuse_a=*/false, /*reuse_b=*/false);
    }

    const float sb = sqB[n0 + l];
    #pragma unroll
    for (int v = 0; v < 8; ++v) {
        const int m = m0 + v + 8 * half;
        float sq = sqA[m] + sb - 2.0f * acc[v];
        float d  = __builtin_sqrtf(fmaxf(sq, 0.0f));
        d = fmaxf(d, 1e-6f);
        dist[(size_t)m * NB + n0 + l] = d;
    }
}

// ---------------------------------------------------------------------------
// Kernel 2 (HBM-bound phase): one wave per triplet; 32 lanes x float4 = 128.
// Edge-row read (512 B) and output write (512 B) fully coalesced per wave.
// edge_reprs is 512 MB streamed once -> non-temporal load; output is 128 MB
// write-once -> non-temporal store. Keeps the 192 MB L2 for dist/emb/ref,
// which are re-read ~256x each across the T triplets.
// ---------------------------------------------------------------------------
__global__ void gp_gather_kernel(const float* __restrict__ emb,
                                 const float* __restrict__ ref,
                                 const int*   __restrict__ anc,
                                 const int*   __restrict__ pos,
                                 const int*   __restrict__ neg,
                                 const float* __restrict__ edge,
                                 const float* __restrict__ javg_p,
                                 const float* __restrict__ dist,
                                 float* __restrict__ out,
                                 int T) {
    const int t = blockIdx.x * 8 + (threadIdx.x >> 5);
    if (t >= T) return;
    const int lane = threadIdx.x & 31;
    const int c    = lane << 2;                     // float4 column

    const int a = anc[t];
    const int p = pos[t];
    const int n = neg[t];

    const float Dap = dist[(size_t)a * NB + p];
    const float Dan = dist[(size_t)a * NB + n];
    const float ratio = Dap / Dan;
    const bool  negm  = (Dap >= Dan);

    const float J = javg_p[0];
    const float x = (J == 0.0f) ? 1.0e6f : J;
    const float scale = __expf(-ALPHA_F / x);

    const v4f e4 = __builtin_nontemporal_load(
        (const v4f*)(edge + ((size_t)a * NB + n) * NBITS + c));
    const v4f zi = *(const v4f*)(emb + (size_t)a * NBITS + c);
    const v4f zj = *(const v4f*)(ref + (size_t)n * NBITS + c);

    v4f o;
    {
        float le, r;
        le = e4.x * scale; r = (1.f - le) * ratio + le;
        o.x = negm ? zj.x : (1.f - r) * zi.x + r * zj.x;
        le = e4.y * scale; r = (1.f - le) * ratio + le;
        o.y = negm ? zj.y : (1.f - r) * zi.y + r * zj.y;
        le = e4.z * scale; r = (1.f - le) * ratio + le;
        o.z = negm ? zj.z : (1.f - r) * zi.z + r * zj.z;
        le = e4.w * scale; r = (1.f - le) * ratio + le;
        o.w = negm ? zj.w : (1.f - r) * zi.w + r * zj.w;
    }
    __builtin_nontemporal_store(o, (v4f*)(out + (size_t)t * NBITS + c));
}

// ---------------------------------------------------------------------------
extern "C" void kernel_launch(void* const* d_in, const int* in_sizes, int n_in,
                              void* d_out, int out_size, void* d_ws, size_t ws_size,
                              hipStream_t stream) {
    const float* emb  = (const float*)d_in[0];   // (1024, 128)
    const float* ref  = (const float*)d_in[1];   // (1024, 128)
    const int*   anc  = (const int*)d_in[2];     // (T,)
    const int*   pos  = (const int*)d_in[3];     // (T,)
    const int*   neg  = (const int*)d_in[4];     // (T,)
    const float* edge = (const float*)d_in[5];   // (1024, 1024, 128)
    const float* javg = (const float*)d_in[6];   // scalar
    float*       out  = (float*)d_out;           // (T, 128)
    const int    T    = in_sizes[2];

    // Workspace layout (floats): dist[1024*1024] | sqA[1024] | sqB[1024]
    float* dist = (float*)d_ws;
    float* sqA  = dist + (size_t)NB * NB;
    float* sqB  = sqA + NB;

    // 0) row norms: 2048 threads
    gp_norms_kernel<<<(2 * NB + 255) / 256, 256, 0, stream>>>(emb, ref, sqA, sqB);

    // 1) distance matrix: 4096 tiles, 8 waves (tiles) per 256-thread block
    gp_dist_wmma_kernel<<<(NB / 16) * (NB / 16) / 8, 256, 0, stream>>>(
        emb, ref, sqA, sqB, dist);

    // 2) triplet gather/blend: one wave per triplet, 8 triplets per block
    gp_gather_kernel<<<(T + 7) / 8, 256, 0, stream>>>(
        emb, ref, anc, pos, neg, edge, javg, dist, out, T);
}